// LocalConvAttention_2559800509204
// MI455X (gfx1250) — compile-verified
//
#include <hip/hip_runtime.h>

typedef float v2f __attribute__((ext_vector_type(2)));
typedef float v8f __attribute__((ext_vector_type(8)));

#define HW 16384   // 128*128 plane
#define CH 128
#define W  128

// ---------------------------------------------------------------------------
// Kernel 1: fused depthwise branches -> add[n][c][h][w]  (written into d_out)
// One workgroup = (16-row block, channel, batch). 128 threads = 4 waves,
// thread j = column. 26 input rows staged in LDS; horizontal passes kept in
// registers (fully unrolled) and reused across 16 output rows.
// ---------------------------------------------------------------------------
__global__ __launch_bounds__(128)
void dwconv_kernel(const float* __restrict__ x,
                   const float* __restrict__ w0,   const float* __restrict__ b0,
                   const float* __restrict__ w0_1, const float* __restrict__ b0_1,
                   const float* __restrict__ w0_2, const float* __restrict__ b0_2,
                   const float* __restrict__ w1_1, const float* __restrict__ b1_1,
                   const float* __restrict__ w1_2, const float* __restrict__ b1_2,
                   float* __restrict__ addout) {
    __shared__ float xt[26][W];

    const int j     = threadIdx.x;        // column 0..127
    const int rblk  = blockIdx.x;         // 0..7
    const int c     = blockIdx.y;         // channel
    const int n     = blockIdx.z;         // batch
    const int rbase = rblk * 16;

    const float* xp = x + ((long)(n * CH + c)) * HW;

    // stage rows rbase-5 .. rbase+20 (zero rows outside image)
    #pragma unroll
    for (int rr = 0; rr < 26; ++rr) {
        int r = rbase - 5 + rr;
        xt[rr][j] = ((unsigned)r < 128u) ? xp[r * W + j] : 0.0f;
    }
    __syncthreads();

    // per-channel weights (uniform across block; L2-hot)
    float W33[9], W01[5], W02[5], W11[11], W12[11];
    #pragma unroll
    for (int i = 0; i < 9; ++i)  W33[i] = w0  [c * 9  + i];
    #pragma unroll
    for (int i = 0; i < 5; ++i)  W01[i] = w0_1[c * 5  + i];
    #pragma unroll
    for (int i = 0; i < 5; ++i)  W02[i] = w0_2[c * 5  + i];
    #pragma unroll
    for (int i = 0; i < 11; ++i) W11[i] = w1_1[c * 11 + i];
    #pragma unroll
    for (int i = 0; i < 11; ++i) W12[i] = w1_2[c * 11 + i];
    const float vb0   = b0  [c];
    const float vb0_1 = b0_1[c];
    const float vb0_2 = b0_2[c];
    const float vb1_1 = b1_1[c];
    const float vb1_2 = b1_2[c];

    // horizontal 11-tap for all 26 rows (image rows rbase-5+rr).
    // Intermediate of the 1x11 conv is (conv + bias) inside the image, 0 outside.
    float h11v[26];
    #pragma unroll
    for (int rr = 0; rr < 26; ++rr) {
        float s = 0.0f;
        #pragma unroll
        for (int t = 0; t < 11; ++t) {
            int col = j + t - 5;
            if ((unsigned)col < 128u) s += W11[t] * xt[rr][col];
        }
        int r = rbase - 5 + rr;
        h11v[rr] = ((unsigned)r < 128u) ? (s + vb1_1) : 0.0f;
    }

    // horizontal 5-tap for rows rbase-2 .. rbase+17 (tile rows 3..22)
    float h5v[20];
    #pragma unroll
    for (int rr = 0; rr < 20; ++rr) {
        float s = 0.0f;
        #pragma unroll
        for (int t = 0; t < 5; ++t) {
            int col = j + t - 2;
            if ((unsigned)col < 128u) s += W01[t] * xt[rr + 3][col];
        }
        int r = rbase - 2 + rr;
        h5v[rr] = ((unsigned)r < 128u) ? (s + vb0_1) : 0.0f;
    }

    float* ap = addout + (((long)(n * CH + c)) * 128 + rbase) * W + j;
    #pragma unroll
    for (int ro = 0; ro < 16; ++ro) {
        // 3x3 branch (zero-padded x; tile rows ro+4..ro+6)
        float a33 = vb0;
        #pragma unroll
        for (int dr = 0; dr < 3; ++dr) {
            #pragma unroll
            for (int dc = 0; dc < 3; ++dc) {
                int col = j + dc - 1;
                if ((unsigned)col < 128u) a33 += W33[dr * 3 + dc] * xt[ro + 4 + dr][col];
            }
        }
        // vertical 5-tap on biased intermediate
        float a15 = vb0_2;
        #pragma unroll
        for (int dr = 0; dr < 5; ++dr) a15 += W02[dr] * h5v[ro + dr];
        // vertical 11-tap on biased intermediate
        float a111 = vb1_2;
        #pragma unroll
        for (int dr = 0; dr < 11; ++dr) a111 += W12[dr] * h11v[ro + dr];

        ap[ro * W] = a33 + a15 + a111;
    }
}

// ---------------------------------------------------------------------------
// Kernel 2: mixer = (add @ w3^T + b3) * x, via V_WMMA_F32_16X16X4_F32.
// One workgroup = 128 consecutive pixels of one image plane, 256 threads.
// LDS: add[128px][132], w3^T[128][129], per-wave 16x17 transpose scratch.
// Wave w owns out-channel stripe [16w,16w+16); loops 8 pixel blocks, K=128
// as 32 k-steps of 4, with A fragments batched 8-at-a-time so one
// s_wait_dscnt covers 8 WMMAs. Runs in-place on d_out (reads fully staged
// before any store; pixel windows are disjoint across workgroups).
// ---------------------------------------------------------------------------
__global__ __launch_bounds__(256)
void mixer_kernel(const float* add,               // aliases out
                  const float* __restrict__ x,
                  const float* __restrict__ w3,
                  const float* __restrict__ b3,
                  float* out) {
    extern __shared__ float lds[];
    float* addl = lds;                         // [128][132]
    float* wt   = lds + 128 * 132;             // [128][129]
    float* scr  = wt  + 128 * 129;             // [8][16*17]

    const int tid  = threadIdx.x;
    const int wave = tid >> 5;
    const int lane = tid & 31;

    const int n   = blockIdx.x >> 7;           // 128 pixel-blocks per image
    const int pin = (blockIdx.x & 127) * 128;  // pixel offset within plane

    const float* addn = add + (long)n * CH * HW + pin;
    const float* xn   = x   + (long)n * CH * HW + pin;
    float*       outn = out + (long)n * CH * HW + pin;

    // stage add[px][c] (coalesced: 128 consecutive px per channel row)
    {
        const int px = tid & 127;
        const int ch = tid >> 7;               // 0 or 1
        #pragma unroll 4
        for (int k = 0; k < 64; ++k) {
            int c = k * 2 + ch;
            addl[px * 132 + c] = addn[(long)c * HW + px];
        }
    }
    // stage w3 transposed: wt[c][o] = w3[o*128+c]
    #pragma unroll 4
    for (int k = 0; k < 64; ++k) {
        int idx = k * 256 + tid;
        int o = idx >> 7, c = idx & 127;
        wt[c * 129 + o] = w3[idx];
    }
    __syncthreads();

    // Preload B fragments (w3^T) for this wave's out-channel stripe.
    // B 4x16 layout: VGPR0 = K rows {0,2}, VGPR1 = K rows {1,3}; N = lane%16.
    const int ob   = wave * 16;
    const int ncol = ob + (lane & 15);
    const int krow = (lane >> 4) * 2;          // 0 (lanes 0-15) / 2 (lanes 16-31)
    v2f B[32];
    #pragma unroll
    for (int kk = 0; kk < 32; ++kk) {
        int kb = kk * 4 + krow;
        B[kk].x = wt[kb * 129 + ncol];
        B[kk].y = wt[(kb + 1) * 129 + ncol];
    }

    // Epilogue biases this lane will need: o = i*2 + (lane>>4), i = 0..7
    float bv[8];
    #pragma unroll
    for (int i = 0; i < 8; ++i) bv[i] = b3[ob + i * 2 + (lane >> 4)];

    const int mrow = lane & 15;                // A row / C column
    float* myscr = scr + wave * (16 * 17);

    for (int pb = 0; pb < 8; ++pb) {
        v8f acc = {};
        const float* arow = &addl[(pb * 16 + mrow) * 132 + krow];
        // A 16x4 layout: lane m<16 holds K0,K1; lane m>=16 holds K2,K3.
        // Load 8 k-steps of A up front, then chain 8 WMMAs: one LDS wait
        // per 8 matrix ops instead of per 2.
        #pragma unroll
        for (int kg = 0; kg < 4; ++kg) {
            v2f A8[8];
            #pragma unroll
            for (int i = 0; i < 8; ++i)
                A8[i] = *(const v2f*)(arow + (kg * 8 + i) * 4);
            #pragma unroll
            for (int i = 0; i < 8; ++i)
                acc = __builtin_amdgcn_wmma_f32_16x16x4_f32(
                    false, A8[i], false, B[kg * 8 + i], (short)0, acc, false, false);
        }

        // C/D layout: VGPR i -> M = i (lanes 0-15) / 8+i (lanes 16-31), N = lane%16.
        // Transpose through LDS for coalesced epilogue.
        const int mbase = (lane < 16) ? 0 : 8;
        #pragma unroll
        for (int i = 0; i < 8; ++i)
            myscr[(mbase + i) * 17 + (lane & 15)] = acc[i];

        // same-wave DS ops are in-order: reads below see the writes above
        #pragma unroll
        for (int i = 0; i < 8; ++i) {
            int o = i * 2 + (lane >> 4);       // 0..15 within stripe
            int p = lane & 15;                 // pixel within block
            float m = myscr[p * 17 + o];
            long off = (long)(ob + o) * HW + pb * 16 + p;
            outn[off] = (m + bv[i]) * xn[off];
        }
    }
}

// ---------------------------------------------------------------------------
extern "C" void kernel_launch(void* const* d_in, const int* in_sizes, int n_in,
                              void* d_out, int out_size, void* d_ws, size_t ws_size,
                              hipStream_t stream) {
    const float* x    = (const float*)d_in[0];
    const float* w0   = (const float*)d_in[1];
    const float* b0   = (const float*)d_in[2];
    const float* w0_1 = (const float*)d_in[3];
    const float* b0_1 = (const float*)d_in[4];
    const float* w0_2 = (const float*)d_in[5];
    const float* b0_2 = (const float*)d_in[6];
    const float* w1_1 = (const float*)d_in[7];
    const float* b1_1 = (const float*)d_in[8];
    const float* w1_2 = (const float*)d_in[9];
    const float* b1_2 = (const float*)d_in[10];
    const float* w3   = (const float*)d_in[11];
    const float* b3   = (const float*)d_in[12];
    float* out = (float*)d_out;

    // Pass 1: depthwise sum -> out (used as 'add' scratch)
    dwconv_kernel<<<dim3(8, 128, 16), 128, 0, stream>>>(
        x, w0, b0, w0_1, b0_1, w0_2, b0_2, w1_1, b1_1, w1_2, b1_2, out);

    // Pass 2: WMMA channel mix + gating, in-place on out
    size_t shmem = (size_t)(128 * 132 + 128 * 129 + 8 * 16 * 17) * sizeof(float);
    mixer_kernel<<<dim3(16 * 128), 256, shmem, stream>>>(out, x, w3, b3, out);
}